// Involution2d_6004364280746
// MI455X (gfx1250) — compile-verified
//
#include <hip/hip_runtime.h>

#define Bn   2
#define Cc   64
#define Hh   128
#define Ww   128
#define HW   (Hh * Ww)          // 16384
#define CrD  16
#define KK   49
#define EPSV 1e-5f

typedef float v2f __attribute__((ext_vector_type(2)));
typedef float v4f __attribute__((ext_vector_type(4)));
typedef float v8f __attribute__((ext_vector_type(8)));
typedef int   v4i __attribute__((vector_size(16)));   // matches builtin param

#define AS1 __attribute__((address_space(1)))
#define AS3 __attribute__((address_space(3)))

#if __has_builtin(__builtin_amdgcn_global_load_async_to_lds_b128) && \
    __has_builtin(__builtin_amdgcn_s_wait_asynccnt)
#define HAVE_ASYNC_LDS 1
#endif

// ---------------------------------------------------------------------------
// Stage 1: reduced[b,o,p] = sum_c w_reduce[o,c] * X[b,c,p] + b_reduce[o]
// GEMM M=16 (o), K=64 (c), N=16-per-wave tile of p.  V_WMMA_F32_16X16X4_F32.
// ---------------------------------------------------------------------------
__global__ __launch_bounds__(256) void inv_reduce_wmma(
    const float* __restrict__ X, const float* __restrict__ Wr,
    const float* __restrict__ br, float* __restrict__ red)
{
    const int wave = blockIdx.x * 8 + (threadIdx.x >> 5);
    const int lane = threadIdx.x & 31;
    const int n    = lane & 15;
    const int half = lane >> 4;
    const int b    = wave >> 10;            // 1024 N-tiles per batch image
    const int p0   = (wave & 1023) << 4;

    const float* Xb = X + (size_t)b * Cc * HW;

    v8f acc = 0.0f;
    #pragma unroll
    for (int k0 = 0; k0 < Cc; k0 += 4) {
        const int c0 = k0 + 2 * half;
        v2f a, bm;
        a.x  = Wr[n * Cc + c0];                      // A 16x4: lane m, K=2*half(+1)
        a.y  = Wr[n * Cc + c0 + 1];
        bm.x = Xb[(size_t)c0 * HW + p0 + n];         // B 4x16: col n, K=2*half(+1)
        bm.y = Xb[(size_t)(c0 + 1) * HW + p0 + n];
        acc = __builtin_amdgcn_wmma_f32_16x16x4_f32(
                  false, a, false, bm, (short)0, acc, false, false);
    }

    float* rb = red + (size_t)b * CrD * HW + p0 + n;
    #pragma unroll
    for (int r = 0; r < 8; ++r) {
        const int m = r + 8 * half;                  // D: VGPR r -> row r+8*half
        rb[(size_t)m * HW] = acc[r] + br[m];
    }
}

// ---------------------------------------------------------------------------
// Stage 2: per-channel batch stats -> folded BN scale/shift.
// ---------------------------------------------------------------------------
__global__ __launch_bounds__(256) void inv_stats(
    const float* __restrict__ red, const float* __restrict__ gamma,
    const float* __restrict__ beta, float* __restrict__ stats)
{
    __shared__ float s_sum[256];
    __shared__ float s_sq[256];
    const int ch  = blockIdx.x;
    const int tid = threadIdx.x;

    float sum = 0.f, sq = 0.f;
    for (int b = 0; b < Bn; ++b) {
        const float* rp = red + ((size_t)b * CrD + ch) * HW;
        for (int i = tid; i < HW; i += 256) {
            const float v = rp[i];
            sum += v; sq += v * v;
        }
    }
    s_sum[tid] = sum; s_sq[tid] = sq;
    __syncthreads();
    for (int s = 128; s > 0; s >>= 1) {
        if (tid < s) { s_sum[tid] += s_sum[tid + s]; s_sq[tid] += s_sq[tid + s]; }
        __syncthreads();
    }
    if (tid == 0) {
        const float N     = (float)(Bn * HW);
        const float mean  = s_sum[0] / N;
        const float var   = s_sq[0] / N - mean * mean;   // population var
        const float scale = gamma[ch] * rsqrtf(var + EPSV);
        stats[ch]       = scale;
        stats[CrD + ch] = beta[ch] - mean * scale;
    }
}

// ---------------------------------------------------------------------------
// Stage 3: kern[b,k,p] = sum_c w_span[k,c] * relu(BN(reduced[b,c,p])) + b_span[k]
// GEMM M=49 (4 masked 16-row tiles), K=16, BN+ReLU fused into B-operand load.
// ---------------------------------------------------------------------------
__global__ __launch_bounds__(256) void inv_span_wmma(
    const float* __restrict__ red, const float* __restrict__ stats,
    const float* __restrict__ Ws, const float* __restrict__ bs,
    float* __restrict__ kern)
{
    const int wave = blockIdx.x * 8 + (threadIdx.x >> 5);
    const int lane = threadIdx.x & 31;
    const int n    = lane & 15;
    const int half = lane >> 4;
    const int b    = wave >> 10;
    const int p    = ((wave & 1023) << 4) + n;

    v2f bv[4];
    const float* rb = red + (size_t)b * CrD * HW + p;
    #pragma unroll
    for (int j = 0; j < 4; ++j) {
        const int c0 = 4 * j + 2 * half;
        const float v0 = rb[(size_t)c0 * HW]       * stats[c0]     + stats[CrD + c0];
        const float v1 = rb[(size_t)(c0 + 1) * HW] * stats[c0 + 1] + stats[CrD + c0 + 1];
        bv[j].x = v0 > 0.f ? v0 : 0.f;
        bv[j].y = v1 > 0.f ? v1 : 0.f;
    }

    float* kb = kern + (size_t)b * KK * HW + p;
    #pragma unroll
    for (int mt = 0; mt < 4; ++mt) {
        const int row = mt * 16 + n;
        v8f acc = 0.0f;
        #pragma unroll
        for (int j = 0; j < 4; ++j) {
            const int c0 = 4 * j + 2 * half;
            v2f a;
            a.x = (row < KK) ? Ws[row * CrD + c0]     : 0.f;
            a.y = (row < KK) ? Ws[row * CrD + c0 + 1] : 0.f;
            acc = __builtin_amdgcn_wmma_f32_16x16x4_f32(
                      false, a, false, bv[j], (short)0, acc, false, false);
        }
        #pragma unroll
        for (int r = 0; r < 8; ++r) {
            const int krow = mt * 16 + r + 8 * half;
            if (krow < KK) kb[(size_t)krow * HW] = acc[r] + bs[krow];
        }
    }
}

// ---------------------------------------------------------------------------
// Stage 4: involution apply.
// Block = (b, h, 8-channel group): kern row tile (49x128 f = 25 KB) staged in
// LDS via async global->LDS (ASYNCcnt), shared by 8 waves (one channel each).
// Lane owns 4 consecutive w; X window read as 3 aligned b128 loads with
// zero-fill at row ends (zero-padding semantics, no per-tap bounds checks).
// ---------------------------------------------------------------------------
__global__ __launch_bounds__(256) void inv_apply(
    const float* __restrict__ X, const float* __restrict__ kern,
    float* __restrict__ out)
{
    __shared__ float skern[KK * Ww];                 // 25088 B

    const int tid = threadIdx.x;
    const int cg  = blockIdx.x & 7;                  // channel group (8 ch)
    const int h   = (blockIdx.x >> 3) & (Hh - 1);
    const int b   = blockIdx.x >> 10;

    // Cooperative stage of kern[b, 0..48, h, :]  (49 rows x 32 float4 chunks)
    const float* kbase = kern + (size_t)b * KK * HW + h * Ww;
    for (int i = tid; i < KK * (Ww / 4); i += 256) {
        const int k   = i >> 5;
        const int seg = (i & 31) << 2;
        const float* gp = kbase + (size_t)k * HW + seg;
        float*       lp = &skern[k * Ww + seg];
#ifdef HAVE_ASYNC_LDS
        __builtin_amdgcn_global_load_async_to_lds_b128(
            (AS1 v4i*)(void*)gp, (AS3 v4i*)(void*)lp, 0, 0);
#else
        *(v4f*)lp = *(const v4f*)gp;
#endif
    }
#ifdef HAVE_ASYNC_LDS
    __builtin_amdgcn_s_wait_asynccnt(0);
#endif
    __syncthreads();

    const int c  = cg * 8 + (tid >> 5);              // one channel per wave
    const int w0 = (tid & 31) << 2;                  // 4 outputs per lane

    const float* Xc = X + ((size_t)b * Cc + c) * HW;
    v4f acc = 0.0f;

    #pragma unroll
    for (int kh = 0; kh < 7; ++kh) {
        const int ih = h + kh - 3;
        if ((unsigned)ih < (unsigned)Hh) {           // wave-uniform branch
            const float* xr = Xc + (size_t)ih * Ww;
            const v4f z = 0.0f;
            float xq[12];                            // window [w0-4 .. w0+8)
            *(v4f*)&xq[0] = (w0 > 0)        ? *(const v4f*)(xr + w0 - 4) : z;
            *(v4f*)&xq[4] =                   *(const v4f*)(xr + w0);
            *(v4f*)&xq[8] = (w0 <= Ww - 8)  ? *(const v4f*)(xr + w0 + 4) : z;
            #pragma unroll
            for (int kw = 0; kw < 7; ++kw) {
                const v4f kv = *(const v4f*)&skern[(kh * 7 + kw) * Ww + w0];
                acc.x += kv.x * xq[1 + kw];          // iw = w0+i+kw-3 -> xq[i+kw+1]
                acc.y += kv.y * xq[2 + kw];
                acc.z += kv.z * xq[3 + kw];
                acc.w += kv.w * xq[4 + kw];
            }
        }
    }
    *(v4f*)(out + ((size_t)b * Cc + c) * HW + h * Ww + w0) = acc;
}

// ---------------------------------------------------------------------------
extern "C" void kernel_launch(void* const* d_in, const int* in_sizes, int n_in,
                              void* d_out, int out_size, void* d_ws, size_t ws_size,
                              hipStream_t stream) {
    (void)in_sizes; (void)n_in; (void)out_size; (void)ws_size;

    const float* X     = (const float*)d_in[0];
    const float* Wr    = (const float*)d_in[1];
    const float* br    = (const float*)d_in[2];
    const float* gamma = (const float*)d_in[3];
    const float* beta  = (const float*)d_in[4];
    const float* Ws    = (const float*)d_in[5];
    const float* bs    = (const float*)d_in[6];
    float* out = (float*)d_out;

    float* ws    = (float*)d_ws;
    float* red   = ws;                                  // B*16*HW floats
    float* stats = ws + (size_t)Bn * CrD * HW;          // 32 floats
    float* kern  = stats + 2 * CrD;                     // B*49*HW floats

    hipLaunchKernelGGL(inv_reduce_wmma, dim3(256), dim3(256), 0, stream,
                       X, Wr, br, red);
    hipLaunchKernelGGL(inv_stats, dim3(CrD), dim3(256), 0, stream,
                       red, gamma, beta, stats);
    hipLaunchKernelGGL(inv_span_wmma, dim3(256), dim3(256), 0, stream,
                       red, stats, Ws, bs, kern);
    // Stage 4: Bn * Hh * (Cc/8) = 2048 blocks
    hipLaunchKernelGGL(inv_apply, dim3(Bn * Hh * (Cc / 8)), dim3(256), 0, stream,
                       X, kern, out);
}